// SubconsciousCore_46660524704457
// MI455X (gfx1250) — compile-verified
//
#include <hip/hip_runtime.h>
#include <hip/hip_bf16.h>

typedef __attribute__((ext_vector_type(2))) float v2f;
typedef __attribute__((ext_vector_type(8))) float v8f;

#define DIM 1024
#define NMEM 131072

__device__ __forceinline__ v8f wmma_f32_16x16x4(v2f a, v2f b, v8f c) {
  // 8 args: (neg_a, A, neg_b, B, c_mod, C, reuse_a, reuse_b)
  return __builtin_amdgcn_wmma_f32_16x16x4_f32(false, a, false, b, (short)0, c, false, false);
}

// ---------------------------------------------------------------------------
// K1: sims[i] = dot(mem[i,:], z) / max(||mem[i,:]||, eps)   (monotone-equal to
// reference cosine sims; only the top-8 ordering is consumed downstream).
// One wave per 16-row tile. A = 16x4 fp32 tile; B = z chunk broadcast to all
// 16 columns, so every C column holds the same 16 dots.
// ---------------------------------------------------------------------------
__global__ void k_sims(const float* __restrict__ mem, const float* __restrict__ z,
                       float* __restrict__ sims, int N, int D) {
  int wave = (int)((blockIdx.x * blockDim.x + threadIdx.x) >> 5);
  int lane = threadIdx.x & 31;
  int row0 = wave * 16;
  if (row0 >= N) return;                 // wave-uniform
  int m    = lane & 15;
  int koff = (lane >> 4) * 2;            // lanes 0-15: K{0,1}; lanes 16-31: K{2,3}
  const float* rowp = mem + (size_t)(row0 + m) * D + koff;
  const float* zp   = z + koff;
  v8f acc = {};
  float ss = 0.f;
  for (int k0 = 0; k0 < D; k0 += 4) {
    if ((k0 & 31) == 0) __builtin_prefetch(rowp + k0 + 128, 0, 3); // next line
    v2f a = *(const v2f*)(rowp + k0);
    v2f b = *(const v2f*)(zp + k0);
    acc = wmma_f32_16x16x4(a, b, acc);
    ss = fmaf(a.x, a.x, ss);
    ss = fmaf(a.y, a.y, ss);
  }
  // norm^2 of row m: lanes l and l+16 cover complementary K positions
  ss += __shfl_xor(ss, 16, 32);
  // dots: C vgpr r, lanes 0-15 hold D[r][*]==dot[r], lanes 16-31 hold dot[r+8]
  float dot = 0.f;
#pragma unroll
  for (int j = 0; j < 8; ++j) {
    float ox = __shfl_xor(acc[j], 16, 32);   // lanes<16 receive dot[j+8]
    dot = (m == j)     ? acc[j] : dot;
    dot = (m == j + 8) ? ox     : dot;
  }
  if (lane < 16) {
    float nrm = fmaxf(sqrtf(ss), 1e-12f);
    sims[row0 + m] = dot / nrm;
  }
}

// ---------------------------------------------------------------------------
// K2: top-8 by repeated block-wide argmax (ties -> smallest index, like
// jax.lax.top_k). Single block, 256 threads; 8 passes over 131072 floats.
// ---------------------------------------------------------------------------
__global__ void k_top8(const float* __restrict__ sims, int* __restrict__ idxTop, int N) {
  __shared__ unsigned long long red[256];
  __shared__ int chosen[8];
  int t = threadIdx.x;
  for (int r = 0; r < 8; ++r) {
    unsigned long long best = 0ull;
    for (int i = t; i < N; i += 256) {
      bool skip = false;
      for (int j = 0; j < r; ++j) skip |= (chosen[j] == i);
      if (skip) continue;
      unsigned u = __float_as_uint(sims[i]);
      u = (u & 0x80000000u) ? ~u : (u | 0x80000000u);        // orderable uint
      unsigned long long key =
          ((unsigned long long)u << 32) | (unsigned)(~(unsigned)i); // tie->low idx
      best = (key > best) ? key : best;
    }
    red[t] = best;
    __syncthreads();
    for (int s = 128; s > 0; s >>= 1) {
      if (t < s) red[t] = (red[t + s] > red[t]) ? red[t + s] : red[t];
      __syncthreads();
    }
    if (t == 0) {
      int win = (int)~((unsigned)(red[0] & 0xFFFFFFFFu));
      chosen[r] = win;
      idxTop[r] = win;
    }
    __syncthreads();
  }
}

// ---------------------------------------------------------------------------
// K3: build padded candidate matrix Cc (16 x D): rows 0..2 = means of top-8/2/3
// protos, rows 3..6 = clipped dreams, rows 7..15 = zero pad for WMMA.
// ---------------------------------------------------------------------------
__global__ void k_build(const float* __restrict__ mem, const int* __restrict__ idxTop,
                        const float* __restrict__ z, const float* __restrict__ eps,
                        float* __restrict__ Cc, int D) {
  int c = blockIdx.x * blockDim.x + threadIdx.x;
  if (c >= D) return;
  float p[8];
#pragma unroll
  for (int j = 0; j < 8; ++j) p[j] = mem[(size_t)idxTop[j] * D + c];
  float s8 = 0.f;
#pragma unroll
  for (int j = 0; j < 8; ++j) s8 += p[j];
  Cc[0 * D + c] = s8 * 0.125f;
  Cc[1 * D + c] = (p[0] + p[1]) * 0.5f;
  Cc[2 * D + c] = (p[0] + p[1] + p[2]) * (1.f / 3.f);
  float zc = z[c];
#pragma unroll
  for (int j = 0; j < 4; ++j) {
    float dv = fmaf(0.08f, eps[j * D + c], zc);
    Cc[(3 + j) * D + c] = fminf(fmaxf(dv, -2.f), 2.f);
  }
#pragma unroll
  for (int r = 7; r < 16; ++r) Cc[r * D + c] = 0.f;
}

// ---------------------------------------------------------------------------
// K4: qv = tanh(query_w @ [z;h] + qb) and t1 = tanh(gate1_w @ [z;h] + g1b).
// One wave per output row (2*D waves total).
// ---------------------------------------------------------------------------
__global__ void k_matvec2(const float* __restrict__ qw, const float* __restrict__ qb,
                          const float* __restrict__ g1w, const float* __restrict__ g1b,
                          const float* __restrict__ z, const float* __restrict__ h,
                          float* __restrict__ qv, float* __restrict__ t1, int D) {
  int wave = (int)((blockIdx.x * blockDim.x + threadIdx.x) >> 5);
  int lane = threadIdx.x & 31;
  const float* W; const float* b; float* out; int row;
  if (wave < D) { W = qw;  b = qb;  out = qv; row = wave; }
  else          { W = g1w; b = g1b; out = t1; row = wave - D; }
  const float* wr = W + (size_t)row * (2 * D);
  float s = 0.f;
  for (int c = lane; c < 2 * D; c += 32) {
    float x = (c < D) ? z[c] : h[c - D];
    s = fmaf(wr[c], x, s);
  }
#pragma unroll
  for (int o = 16; o > 0; o >>= 1) s += __shfl_xor(s, o, 32);
  if (lane == 0) out[row] = tanhf(s + b[row]);
}

// ---------------------------------------------------------------------------
// K5: A = tanh(Cc @ cand_w^T + cand_b): (16x1024) x (1024x1024) fp32 WMMA GEMM.
// One wave per 16-column output tile (64 waves). B columns = cand_w rows.
// ---------------------------------------------------------------------------
__global__ void k_cand(const float* __restrict__ Cc, const float* __restrict__ Wc,
                       const float* __restrict__ bc, float* __restrict__ Aout, int D) {
  int wave = (int)((blockIdx.x * blockDim.x + threadIdx.x) >> 5);
  int lane = threadIdx.x & 31;
  int n0 = wave * 16;
  if (n0 >= D) return;                    // wave-uniform
  int idx  = lane & 15;
  int koff = (lane >> 4) * 2;
  const float* ap = Cc + (size_t)idx * D + koff;            // A row m = idx
  const float* bp = Wc + (size_t)(n0 + idx) * D + koff;     // B col n = idx
  v8f acc = {};
  for (int k0 = 0; k0 < D; k0 += 4) {
    v2f a = *(const v2f*)(ap + k0);
    v2f b = *(const v2f*)(bp + k0);
    acc = wmma_f32_16x16x4(a, b, acc);
  }
  int mbase = (lane < 16) ? 0 : 8;        // C: lanes16-31 hold M=r+8
  int n = n0 + idx;
  float bias = bc[n];
#pragma unroll
  for (int r = 0; r < 8; ++r) {
    int mm = mbase + r;
    if (mm < 7) Aout[(size_t)mm * D + n] = tanhf(acc[r] + bias);
  }
}

// ---------------------------------------------------------------------------
// K6: scores -> softmax alpha (also written to out), mix_vec = alpha @ Cc,
// and g = sigmoid(gate2_w . t1 + g2b) (computed by wave 7). Single block.
// ---------------------------------------------------------------------------
__global__ void k_attn(const float* __restrict__ Aw, const float* __restrict__ qv,
                       const float* __restrict__ sw, const float* __restrict__ sb,
                       const float* __restrict__ Cc,
                       const float* __restrict__ g2w, const float* __restrict__ g2b,
                       const float* __restrict__ t1,
                       float* __restrict__ mix, float* __restrict__ gsc,
                       float* __restrict__ alphaOut, int D) {
  __shared__ float sc[8];
  __shared__ float al[8];
  int t = threadIdx.x;
  int wave = t >> 5, lane = t & 31;
  if (wave < 7) {
    float s = 0.f;
    for (int d = lane; d < D; d += 32) s = fmaf(Aw[(size_t)wave * D + d] * qv[d], sw[d], s);
#pragma unroll
    for (int o = 16; o > 0; o >>= 1) s += __shfl_xor(s, o, 32);
    if (lane == 0) sc[wave] = s + sb[0];
  } else {
    float s = 0.f;
    for (int d = lane; d < D; d += 32) s = fmaf(g2w[d], t1[d], s);
#pragma unroll
    for (int o = 16; o > 0; o >>= 1) s += __shfl_xor(s, o, 32);
    if (lane == 0) gsc[0] = 1.f / (1.f + expf(-(s + g2b[0])));
  }
  __syncthreads();
  if (t == 0) {
    float mx = sc[0];
    for (int j = 1; j < 7; ++j) mx = fmaxf(mx, sc[j]);
    float den = 0.f, e[7];
    for (int j = 0; j < 7; ++j) { e[j] = expf(sc[j] - mx); den += e[j]; }
    for (int j = 0; j < 7; ++j) { al[j] = e[j] / den; alphaOut[j] = al[j]; }
  }
  __syncthreads();
  for (int d = t; d < D; d += blockDim.x) {
    float s = 0.f;
#pragma unroll
    for (int j = 0; j < 7; ++j) s = fmaf(al[j], Cc[(size_t)j * D + d], s);
    mix[d] = s;
  }
}

// ---------------------------------------------------------------------------
// K7: s_t[r] = g * tanh(mix_w[r,:] @ mix_vec + mix_b[r]). One wave per row.
// ---------------------------------------------------------------------------
__global__ void k_final(const float* __restrict__ mw, const float* __restrict__ mb,
                        const float* __restrict__ mix, const float* __restrict__ gsc,
                        float* __restrict__ sOut, int D) {
  int wave = (int)((blockIdx.x * blockDim.x + threadIdx.x) >> 5);
  int lane = threadIdx.x & 31;
  if (wave >= D) return;
  const float* wr = mw + (size_t)wave * D;
  float s = 0.f;
  for (int c = lane; c < D; c += 32) s = fmaf(wr[c], mix[c], s);
#pragma unroll
  for (int o = 16; o > 0; o >>= 1) s += __shfl_xor(s, o, 32);
  if (lane == 0) sOut[wave] = gsc[0] * tanhf(s + mb[wave]);
}

extern "C" void kernel_launch(void* const* d_in, const int* in_sizes, int n_in,
                              void* d_out, int out_size, void* d_ws, size_t ws_size,
                              hipStream_t stream) {
  const float* z    = (const float*)d_in[0];
  const float* h    = (const float*)d_in[1];
  const float* mem  = (const float*)d_in[2];
  const float* eps  = (const float*)d_in[3];
  const float* qw   = (const float*)d_in[4];
  const float* qb   = (const float*)d_in[5];
  const float* cw   = (const float*)d_in[6];
  const float* cb   = (const float*)d_in[7];
  const float* sw   = (const float*)d_in[8];
  const float* sb   = (const float*)d_in[9];
  const float* mw   = (const float*)d_in[10];
  const float* mb   = (const float*)d_in[11];
  const float* g1w  = (const float*)d_in[12];
  const float* g1b  = (const float*)d_in[13];
  const float* g2w  = (const float*)d_in[14];
  const float* g2b  = (const float*)d_in[15];

  const int D = in_sizes[0];          // 1024
  const int N = in_sizes[2] / D;      // 131072

  float* out  = (float*)d_out;        // [0,D): s_t ; [D,D+7): alpha

  // Workspace layout (floats). sims region is reused post-top8.
  float* ws   = (float*)d_ws;
  int*   idxT = (int*)ws;             // 8 ints
  float* g    = ws + 8;               // 1 float
  float* sims = ws + 16;              // N floats (K1,K2 only)
  float* Cc   = ws + 16;              // 16*D (reuses sims region after K2)
  float* Amat = Cc + 16 * D;          // 16*D (only 7 rows used)
  float* qv   = Amat + 16 * D;        // D
  float* t1   = qv + D;               // D
  float* mixv = t1 + D;               // D

  // K1: 8192 waves = 1024 blocks x 8 waves
  k_sims<<<N / 128, 256, 0, stream>>>(mem, z, sims, N, D);
  // K2: single-block top-8
  k_top8<<<1, 256, 0, stream>>>(sims, idxT, N);
  // K3: candidate matrix
  k_build<<<(D + 255) / 256, 256, 0, stream>>>(mem, idxT, z, eps, Cc, D);
  // K4: qv and t1 matvecs (2*D waves)
  k_matvec2<<<(2 * D) / 8, 256, 0, stream>>>(qw, qb, g1w, g1b, z, h, qv, t1, D);
  // K5: WMMA GEMM for candidate projections (D/16 waves)
  k_cand<<<D / 128, 256, 0, stream>>>(Cc, cw, cb, Amat, D);
  // K6: scores/softmax/mix_vec/gate
  k_attn<<<1, 256, 0, stream>>>(Amat, qv, sw, sb, Cc, g2w, g2b, t1, mixv, g, out + D, D);
  // K7: final gated output (D waves)
  k_final<<<D / 8, 256, 0, stream>>>(mw, mb, mixv, g, out, D);
}